// StaticRecurrentEntNet_22342419874073
// MI455X (gfx1250) — compile-verified
//
#include <hip/hip_runtime.h>
#include <hip/hip_bf16.h>

typedef __attribute__((ext_vector_type(2))) float v2f;
typedef __attribute__((ext_vector_type(8))) float v8f;

#define B_  128
#define S_  20
#define L_  30
#define E_  32
#define D_  512

// ---------------------------------------------------------------------------
// M = U + V   (512x512)
// ---------------------------------------------------------------------------
__global__ void add_uv_kernel(const float* __restrict__ U, const float* __restrict__ V,
                              float* __restrict__ M) {
  int i = blockIdx.x * blockDim.x + threadIdx.x;
  if (i < D_ * D_) M[i] = U[i] + V[i];
}

// ---------------------------------------------------------------------------
// enc[b,s,:] = sum_l emb[prgrph[b,s,l],:] * mask ; smask[b,s] = any(mask)
// one block per (b,s) row; 128 threads x float4 = 512 floats
// ---------------------------------------------------------------------------
__global__ void encode_kernel(const float* __restrict__ emb,
                              const int* __restrict__ prg,
                              const unsigned char* __restrict__ msk,
                              float* __restrict__ enc,
                              float* __restrict__ smask) {
  int bs  = blockIdx.x;          // 0 .. B*S-1
  int tid = threadIdx.x;         // 0 .. 127
  const int*           ridx = prg + (size_t)bs * L_;
  const unsigned char* rm   = msk + (size_t)bs * L_;
  float4 acc = make_float4(0.f, 0.f, 0.f, 0.f);
  for (int l = 0; l < L_; ++l) {
    if (rm[l]) {
      const float4* e = (const float4*)(emb + (size_t)ridx[l] * D_);
      float4 v = e[tid];
      acc.x += v.x; acc.y += v.y; acc.z += v.z; acc.w += v.w;
    }
  }
  ((float4*)(enc + (size_t)bs * D_))[tid] = acc;
  if (tid == 0) {
    int any = 0;
    for (int l = 0; l < L_; ++l) any |= rm[l];
    smask[bs] = any ? 1.0f : 0.0f;
  }
}

// ---------------------------------------------------------------------------
// WMMA core: 16-row A panel (LDS, row-major 16x512) times 512x512 row-major B.
// Wave `wave` accumulates 4 N-tiles (64 cols). f32 16x16x4 WMMA.
// A layout (ISA 7.12.2): lanes 0-15 -> rows, vgpr pair = (k0,k0+1);
//                        lanes 16-31 -> rows, (k0+2,k0+3).
// B layout assumed mirrored: v0 = K0(lo-lanes)|K2(hi-lanes), v1 = K1|K3,
// cols striped across the 16 lanes of each half.
// ---------------------------------------------------------------------------
__device__ __forceinline__ void wmma_panel_16xD(const float* __restrict__ Hs,
                                                const float* __restrict__ Bmat,
                                                int wave, int lane, v8f acc[4]) {
  const int laneHalf = lane >> 4;     // 0 or 1
  const int ln       = lane & 15;
  for (int k0 = 0; k0 < D_; k0 += 4) {
    const int ka = k0 + laneHalf * 2;
    v2f a;
    a.x = Hs[ln * D_ + ka];
    a.y = Hs[ln * D_ + ka + 1];
#pragma unroll
    for (int ti = 0; ti < 4; ++ti) {
      const int n = (wave * 4 + ti) * 16 + ln;
      const float* Bk = Bmat + (size_t)ka * D_ + n;
      v2f b;
      b.x = Bk[0];
      b.y = Bk[D_];
      acc[ti] = __builtin_amdgcn_wmma_f32_16x16x4_f32(
          /*neg_a=*/false, a, /*neg_b=*/false, b,
          /*c_mod=*/(short)0, acc[ti], /*reuse_a=*/false, /*reuse_b=*/false);
    }
  }
}

// ---------------------------------------------------------------------------
// encW = enc @ W :  (B*S x 512) @ (512 x 512).  One block per 16 rows.
// ---------------------------------------------------------------------------
__global__ void __launch_bounds__(256)
gemm_encw_kernel(const float* __restrict__ enc, const float* __restrict__ W,
                 float* __restrict__ encW) {
  __shared__ float Hs[16 * D_];
  const int r0  = blockIdx.x * 16;
  const int tid = threadIdx.x;
  const float* src = enc + (size_t)r0 * D_;
  for (int i = tid; i < 16 * D_ / 4; i += 256)
    ((float4*)Hs)[i] = ((const float4*)src)[i];
  __syncthreads();

  const int wave = tid >> 5, lane = tid & 31;
  const int laneHalf = lane >> 4, ln = lane & 15;
  v8f acc[4] = {};
  wmma_panel_16xD(Hs, W, wave, lane, acc);

#pragma unroll
  for (int ti = 0; ti < 4; ++ti) {
    const int n = (wave * 4 + ti) * 16 + ln;
#pragma unroll
    for (int i = 0; i < 8; ++i) {
      const int row = i + laneHalf * 8;   // C/D layout: vgpr i -> rows i, i+8
      encW[(size_t)(r0 + row) * D_ + n] = acc[ti][i];
    }
  }
}

// ---------------------------------------------------------------------------
// One recurrence step. Block = 16 rows of hiddens (one b, half its entities).
//   gate  = sigmoid( enc_t . (h + key) ) * m_t
//   htil  = relu( h @ M + encW_t )
//   hnew  = l2norm( h + gate * htil )
// ---------------------------------------------------------------------------
__global__ void __launch_bounds__(256)
step_kernel(const float* __restrict__ Hin, float* __restrict__ Hout,
            const float* __restrict__ keys, const float* __restrict__ M,
            const float* __restrict__ enc, const float* __restrict__ encW,
            const float* __restrict__ smask, int t) {
  __shared__ float Hs[16 * D_];
  __shared__ float red[16][16];
  __shared__ float gg[16];
  __shared__ float ssq[16];

  const int b   = blockIdx.x >> 1;
  const int e0  = (blockIdx.x & 1) * 16;
  const int tid = threadIdx.x;

  const float* h      = Hin  + ((size_t)b * E_ + e0) * D_;
  const float* kk     = keys + ((size_t)b * E_ + e0) * D_;
  const float* enc_t  = enc  + ((size_t)b * S_ + t) * D_;
  const float* encW_t = encW + ((size_t)b * S_ + t) * D_;
  const float  m_t    = smask[b * S_ + t];

  // stage the 16x512 A panel into LDS
  for (int i = tid; i < 16 * D_ / 4; i += 256)
    ((float4*)Hs)[i] = ((const float4*)h)[i];
  if (tid < 16) ssq[tid] = 0.0f;
  __syncthreads();

  // ---- gate logits: 16 rows x 16 partial lanes, each sums 32 strided d ----
  {
    const int row = tid >> 4, sub = tid & 15;
    const float* hr = Hs + row * D_;
    const float* kr = kk + (size_t)row * D_;
    float p = 0.0f;
    for (int d = sub; d < D_; d += 16) p += enc_t[d] * (hr[d] + kr[d]);
    red[row][sub] = p;
  }
  __syncthreads();
  if (tid < 16) {
    float s = 0.0f;
#pragma unroll
    for (int j = 0; j < 16; ++j) s += red[tid][j];
    gg[tid] = m_t * (1.0f / (1.0f + expf(-s)));   // mask folded into gate
  }
  __syncthreads();

  // ---- WMMA GEMM: h @ (U+V) ----
  const int wave = tid >> 5, lane = tid & 31;
  const int laneHalf = lane >> 4, ln = lane & 15;
  v8f acc[4] = {};
  wmma_panel_16xD(Hs, M, wave, lane, acc);

  // ---- fuse bias + relu + gate + residual; accumulate row sum-of-squares ----
  float hnew[4][8];
  float part[8];
#pragma unroll
  for (int i = 0; i < 8; ++i) part[i] = 0.0f;

#pragma unroll
  for (int ti = 0; ti < 4; ++ti) {
    const int n    = (wave * 4 + ti) * 16 + ln;
    const float bias = encW_t[n];
#pragma unroll
    for (int i = 0; i < 8; ++i) {
      const int row = i + laneHalf * 8;
      float x = acc[ti][i] + bias;
      x = fmaxf(x, 0.0f);                       // relu
      float hn = Hs[row * D_ + n] + gg[row] * x;
      hnew[ti][i] = hn;
      part[i] += hn * hn;
    }
  }
#pragma unroll
  for (int i = 0; i < 8; ++i)
    atomicAdd(&ssq[i + laneHalf * 8], part[i]); // ds_add_f32
  __syncthreads();

  // ---- normalize + store ----
#pragma unroll
  for (int ti = 0; ti < 4; ++ti) {
    const int n = (wave * 4 + ti) * 16 + ln;
#pragma unroll
    for (int i = 0; i < 8; ++i) {
      const int row = i + laneHalf * 8;
      const float rs = rsqrtf(fmaxf(ssq[row], 1e-12f));
      Hout[((size_t)b * E_ + e0 + row) * D_ + n] = hnew[ti][i] * rs;
    }
  }
}

// ---------------------------------------------------------------------------
extern "C" void kernel_launch(void* const* d_in, const int* in_sizes, int n_in,
                              void* d_out, int out_size, void* d_ws, size_t ws_size,
                              hipStream_t stream) {
  const float*         emb   = (const float*)d_in[0];   // (V, D)
  const float*         keys  = (const float*)d_in[1];   // (B, E, D)
  const float*         U     = (const float*)d_in[2];   // (D, D)
  const float*         Vm    = (const float*)d_in[3];   // (D, D)
  const float*         W     = (const float*)d_in[4];   // (D, D)
  const int*           prg   = (const int*)d_in[5];     // (B, S, L)
  const unsigned char* pmask = (const unsigned char*)d_in[6]; // (B, S, L) bool
  float* out = (float*)d_out;                            // (B, E, D)

  // workspace carve (floats): M | enc | encW | smask | Hbuf  (~20 MB)
  float* ws    = (float*)d_ws;
  float* M     = ws;                                  // 512*512      = 262144
  float* enc   = M     + (size_t)D_ * D_;             // B*S*D        = 1310720
  float* encW  = enc   + (size_t)B_ * S_ * D_;        // B*S*D        = 1310720
  float* smask = encW  + (size_t)B_ * S_ * D_;        // B*S          = 2560
  float* Hbuf  = smask + (size_t)B_ * S_;             // B*E*D        = 2097152

  // H0 = entity_keys (re-initialized every call; deterministic)
  hipMemcpyAsync(Hbuf, keys, (size_t)B_ * E_ * D_ * sizeof(float),
                 hipMemcpyDeviceToDevice, stream);

  // M = U + V
  add_uv_kernel<<<(D_ * D_ + 255) / 256, 256, 0, stream>>>(U, Vm, M);

  // enc + sentence mask
  encode_kernel<<<B_ * S_, 128, 0, stream>>>(emb, prg, pmask, enc, smask);

  // encW = enc @ W   (2560 rows -> 160 blocks of 16)
  gemm_encw_kernel<<<(B_ * S_) / 16, 256, 0, stream>>>(enc, W, encW);

  // 20 sequential recurrence steps; each block owns disjoint rows -> in-place.
  for (int t = 0; t < S_; ++t) {
    float* dst = (t == S_ - 1) ? out : Hbuf;
    step_kernel<<<(B_ * E_) / 16, 256, 0, stream>>>(Hbuf, dst, keys, M, enc,
                                                    encW, smask, t);
  }
}